// MultiBoxEL_52802327937557
// MI455X (gfx1250) — compile-verified
//
#include <hip/hip_runtime.h>
#include <hip/hip_bf16.h>
#include <stdint.h>

// Problem constants (fixed by the reference)
#define NC 50000
#define ND 64
#define NK 2
#define NDH 32
#define NP 2048
#define NB 512
#define MC_MULT (4.0f / 2048.0f)
#define LEPS 1e-9f

#define BATCH_TILE 8     // batch items per block
#define POINT_TILE 256   // points per block (== blockDim.x)
#define NSLOTS (BATCH_TILE * 2)        // 16 class rows per block
#define ROW_F 128                       // floats per class row

__global__ __launch_bounds__(256) void multibox_partial_kernel(
    const float* __restrict__ class_weight,   // [C, 128]
    const float* __restrict__ mc_points,      // [P, 32]
    const int*   __restrict__ nf1,            // [B, 2] flat
    float* __restrict__ partials)             // [gridDim.y * gridDim.x]
{
    __shared__ __align__(16) float raw[NSLOTS * ROW_F];        // 16 rows x 128 f = 8KB
    __shared__ __align__(16) float prm[NSLOTS * NK * NDH * 2]; // {mn,mx} pairs = 8KB
    __shared__ float red[POINT_TILE];                          // 1KB

    const int t = threadIdx.x;

    // ---- stage 1 (issue FIRST; it gates the first barrier): async-gather
    // 16 class rows (512B each) into LDS. 512 x 16B chunks; 2 per lane.
    // Per-lane global addr + per-lane LDS addr -> hardware gather via
    // GLOBAL_LOAD_ASYNC_TO_LDS_B128, tracked on ASYNCcnt.
    {
        const int slot0 = t >> 5;                 // chunks t and t+256
        const int slot1 = (t + POINT_TILE) >> 5;
        const int cls0 = nf1[blockIdx.y * NSLOTS + slot0];
        const int cls1 = nf1[blockIdx.y * NSLOTS + slot1];
        const int chunk0 = t & 31;
        const int chunk1 = (t + POINT_TILE) & 31;
        const float* g0 = class_weight + (size_t)cls0 * ROW_F + chunk0 * 4;
        const float* g1 = class_weight + (size_t)cls1 * ROW_F + chunk1 * 4;
        uint32_t l0 = (uint32_t)(uintptr_t)(&raw[slot0 * ROW_F + chunk0 * 4]);
        uint32_t l1 = (uint32_t)(uintptr_t)(&raw[slot1 * ROW_F + chunk1 * 4]);
        asm volatile("global_load_async_to_lds_b128 %0, %1, off"
                     :: "v"(l0), "v"(g0) : "memory");
        asm volatile("global_load_async_to_lds_b128 %0, %1, off"
                     :: "v"(l1), "v"(g1) : "memory");
    }

    // ---- load this thread's point (32 floats) into VGPRs (overlaps asyncs) ----
    const int p = blockIdx.x * POINT_TILE + t;
    float ptf[32];
    {
        const float4* mp4 = (const float4*)(mc_points + (size_t)p * NDH);
        #pragma unroll
        for (int i = 0; i < 8; ++i) {
            float4 v = mp4[i];
            ptf[4*i+0] = v.x; ptf[4*i+1] = v.y; ptf[4*i+2] = v.z; ptf[4*i+3] = v.w;
        }
    }

    asm volatile("s_wait_asynccnt 0x0" ::: "memory");
    __syncthreads();

    // ---- stage 2: rows -> interleaved {mn, mx} pairs in LDS ----
    // 16 slots * 2 boxes * 32 dims = 1024 pairs; 4 per thread.
    #pragma unroll
    for (int q = t; q < NSLOTS * NK * NDH; q += POINT_TILE) {
        const int slot = q >> 6;
        const int r    = q & 63;
        const int k    = r >> 5;
        const int j    = r & 31;
        const float c = raw[slot * ROW_F + k * ND + j];
        const float o = fabsf(raw[slot * ROW_F + k * ND + NDH + j]);
        prm[((slot * NK + k) * NDH + j) * 2 + 0] = c - o;
        prm[((slot * NK + k) * NDH + j) * 2 + 1] = c + o;
    }
    __syncthreads();

    // ---- compute: 8 batch items against this thread's point ----
    float local = 0.0f;
    #pragma unroll
    for (int bi = 0; bi < BATCH_TILE; ++bi) {
        float incs[2];
        #pragma unroll
        for (int s = 0; s < 2; ++s) {
            const int slot = bi * 2 + s;
            float mbest = -3.4e38f;
            #pragma unroll
            for (int k = 0; k < NK; ++k) {
                const float4* bp = (const float4*)&prm[(slot * NK + k) * NDH * 2];
                // two independent min accumulators -> half the serial fmin chain
                float mA = 3.4e38f, mB = 3.4e38f;
                #pragma unroll
                for (int j4 = 0; j4 < 16; ++j4) {   // ds_load_b128, broadcast read
                    const float4 v = bp[j4];
                    const float p0 = ptf[2*j4], p1 = ptf[2*j4+1];
                    mA = fminf(mA, fminf(p0 - v.x, v.y - p0));
                    mB = fminf(mB, fminf(p1 - v.z, v.w - p1));
                }
                mbest = fmaxf(mbest, fminf(mA, mB));
            }
            // max_k sigmoid(m_k) == sigmoid(max_k m_k)  (monotonic)
            incs[s] = 1.0f / (1.0f + __expf(-mbest));
        }
        const float c_inc = incs[0];
        const float d_inc = incs[1];
        const float area1 = fmaxf(c_inc - 0.5f, 0.0f) * MC_MULT;
        const float inter = fmaxf(0.5f * (c_inc + d_inc) - 0.5f, 0.0f) * MC_MULT;
        local += 1.0f - inter / (area1 + LEPS);
    }

    // ---- deterministic block reduction ----
    red[t] = local;
    __syncthreads();
    #pragma unroll
    for (int s = POINT_TILE / 2; s > 0; s >>= 1) {
        if (t < s) red[t] += red[t + s];
        __syncthreads();
    }
    if (t == 0) partials[blockIdx.y * gridDim.x + blockIdx.x] = red[0];
}

__global__ __launch_bounds__(512) void multibox_reduce_kernel(
    const float* __restrict__ partials, float* __restrict__ out, int n)
{
    __shared__ float red[512];
    const int t = threadIdx.x;
    red[t] = (t < n) ? partials[t] : 0.0f;
    __syncthreads();
    #pragma unroll
    for (int s = 256; s > 0; s >>= 1) {
        if (t < s) red[t] += red[t + s];
        __syncthreads();
    }
    if (t == 0) {
        const float loss = red[0] * (1.0f / ((float)NB * (float)NP));
        out[0] = fmaxf(loss, 0.0f);
    }
}

extern "C" void kernel_launch(void* const* d_in, const int* in_sizes, int n_in,
                              void* d_out, int out_size, void* d_ws, size_t ws_size,
                              hipStream_t stream) {
    (void)in_sizes; (void)n_in; (void)out_size; (void)ws_size;
    const float* class_weight = (const float*)d_in[0];
    const float* mc_points    = (const float*)d_in[1];
    const int*   nf1          = (const int*)d_in[2];
    float* out      = (float*)d_out;
    float* partials = (float*)d_ws;

    dim3 grid(NP / POINT_TILE, NB / BATCH_TILE);   // 8 x 64 = 512 blocks
    dim3 block(POINT_TILE);                         // 256 threads = 8 wave32
    multibox_partial_kernel<<<grid, block, 0, stream>>>(class_weight, mc_points, nf1, partials);

    const int nPartials = (NP / POINT_TILE) * (NB / BATCH_TILE); // 512
    multibox_reduce_kernel<<<1, 512, 0, stream>>>(partials, out, nPartials);
}